// AttentionANEWraperChannelsFirstWithCache_14053132992644
// MI455X (gfx1250) — compile-verified
//
#include <hip/hip_runtime.h>

typedef __attribute__((ext_vector_type(2))) float v2f;
typedef __attribute__((ext_vector_type(8))) float v8f;

#define H_    28
#define KV_   4
#define HD_   128
#define G_    7          // H_/KV_
#define LI_   5
#define NL_   16
#define S_    4096
#define D_    3584
#define L_    512
#define SCALE_ 0.08838834764831845f   // 1/sqrt(128)

// -------------------------------------------------------------------------
// Generic fp32 WMMA GEMM:  Y(M,N) = W(M,Kd) @ X(Kd,N) (+ bias(M))
// Block: 256 thr = 8 waves (2 rows x 4 cols).
// Wave tile 64x32 (4x2 WMMA accumulators) -> 8 WMMAs per K-step vs 6 v2f loads.
// Block tile 128 x 128. All dims assumed divisible (they are here).
// -------------------------------------------------------------------------
__global__ __launch_bounds__(256) void gemm_wmma_f32(
    const float* __restrict__ W, const float* __restrict__ bias,
    const float* __restrict__ X, float* __restrict__ Y,
    int M, int Kd, int N)
{
    const int lane = threadIdx.x & 31;
    const int wid  = threadIdx.x >> 5;
    const int wr   = wid & 1;
    const int wc   = wid >> 1;
    const int m0   = blockIdx.x * 128 + wr * 64;
    const int n0   = blockIdx.y * 128 + wc * 32;
    const int half = lane >> 4;      // 0: K={0,1}, 1: K={2,3}
    const int lx   = lane & 15;
    const int koff = half * 2;

    v8f c[4][2];
#pragma unroll
    for (int i = 0; i < 4; ++i)
#pragma unroll
        for (int j = 0; j < 2; ++j) c[i][j] = (v8f){};

    const float* wrow[4];
#pragma unroll
    for (int i = 0; i < 4; ++i)
        wrow[i] = W + (size_t)(m0 + i * 16 + lx) * Kd;
    const float* xcol[2];
#pragma unroll
    for (int j = 0; j < 2; ++j)
        xcol[j] = X + (n0 + j * 16 + lx);

#pragma unroll 2
    for (int k = 0; k < Kd; k += 4) {
        const int kk = k + koff;
        v2f a[4], b[2];
#pragma unroll
        for (int i = 0; i < 4; ++i) {
            a[i][0] = wrow[i][kk];
            a[i][1] = wrow[i][kk + 1];
        }
#pragma unroll
        for (int j = 0; j < 2; ++j) {
            b[j][0] = xcol[j][(size_t)kk * N];
            b[j][1] = xcol[j][(size_t)(kk + 1) * N];
        }
#pragma unroll
        for (int i = 0; i < 4; ++i)
#pragma unroll
            for (int j = 0; j < 2; ++j)
                c[i][j] = __builtin_amdgcn_wmma_f32_16x16x4_f32(
                    false, a[i], false, b[j], (short)0, c[i][j], false, false);
    }

#pragma unroll
    for (int i = 0; i < 4; ++i) {
#pragma unroll
        for (int r = 0; r < 8; ++r) {
            const int m = m0 + i * 16 + r + half * 8;  // C layout: VGPR r -> M=r (lo half), M=r+8 (hi half)
            const float bi = bias ? bias[m] : 0.0f;
#pragma unroll
            for (int j = 0; j < 2; ++j)
                Y[(size_t)m * N + n0 + j * 16 + lx] = c[i][j][r] + bi;
        }
    }
}

// -------------------------------------------------------------------------
// RoPE on Q, in place.  q layout (H, HD, L); cos_t/sin_t layout (HD, L).
// One thread per (h, d<64, l) pair.
// -------------------------------------------------------------------------
__global__ __launch_bounds__(256) void rope_q_kernel(
    float* __restrict__ q, const float* __restrict__ cost,
    const float* __restrict__ sint)
{
    const int idx = blockIdx.x * 256 + threadIdx.x;     // H_*64*L_ total
    const int l = idx % L_;
    const int d = (idx / L_) & 63;
    const int h = idx / (L_ * 64);

    const size_t i1 = ((size_t)h * HD_ + d) * L_ + l;
    const size_t i2 = ((size_t)h * HD_ + d + 64) * L_ + l;
    const float x1 = q[i1];
    const float x2 = q[i2];
    const float c1 = cost[(size_t)d * L_ + l];
    const float s1 = sint[(size_t)d * L_ + l];
    const float c2 = cost[(size_t)(d + 64) * L_ + l];
    const float s2 = sint[(size_t)(d + 64) * L_ + l];
    q[i1] = x1 * c1 - x2 * s1;
    q[i2] = x2 * c2 + x1 * s2;
}

// -------------------------------------------------------------------------
// RoPE on K + transpose: ktmp (KV*HD, L) -> kr (KV, L, HD); cos/sin (L, HD)
// -------------------------------------------------------------------------
__global__ __launch_bounds__(256) void rope_k_kernel(
    const float* __restrict__ ktmp, const float* __restrict__ cosk,
    const float* __restrict__ sink, float* __restrict__ kr)
{
    const int idx = blockIdx.x * 256 + threadIdx.x;     // KV_*64*L_ total
    const int l  = idx % L_;
    const int d  = (idx / L_) & 63;
    const int kv = idx / (L_ * 64);

    const float x1 = ktmp[((size_t)kv * HD_ + d) * L_ + l];
    const float x2 = ktmp[((size_t)kv * HD_ + d + 64) * L_ + l];
    const float c1 = cosk[(size_t)l * HD_ + d];
    const float s1 = sink[(size_t)l * HD_ + d];
    const float c2 = cosk[(size_t)l * HD_ + d + 64];
    const float s2 = sink[(size_t)l * HD_ + d + 64];
    kr[((size_t)kv * L_ + l) * HD_ + d]      = x1 * c1 - x2 * s1;
    kr[((size_t)kv * L_ + l) * HD_ + d + 64] = x2 * c2 + x1 * s2;
}

// -------------------------------------------------------------------------
// V transpose: vtmp (KV*HD, L) -> vt (KV, L, HD)
// -------------------------------------------------------------------------
__global__ __launch_bounds__(256) void transpose_v_kernel(
    const float* __restrict__ vtmp, float* __restrict__ vt)
{
    const int idx = blockIdx.x * 256 + threadIdx.x;     // KV_*HD_*L_ total
    const int l  = idx % L_;
    const int d  = (idx / L_) % HD_;
    const int kv = idx / (L_ * HD_);
    vt[((size_t)kv * L_ + l) * HD_ + d] = vtmp[((size_t)kv * HD_ + d) * L_ + l];
}

// -------------------------------------------------------------------------
// GQA attention. Grid (H, L/16); 256 threads = 8 waves.
// LDS holds the full 4096x16 score tile (256KB; CDNA5 WGP has 320KB).
// Fresh K/V rows (positions [cp, cp+L)) come from workspace; everything
// else comes from the (unmodified) input caches.
// -------------------------------------------------------------------------
__global__ __launch_bounds__(256) void attn_kernel(
    const float* __restrict__ q,       // (H, HD, L) — RoPE'd
    const float* __restrict__ krope,   // (KV, L, HD)
    const float* __restrict__ vt,      // (KV, L, HD)
    const float* __restrict__ kcache,  // (NL, KV, S, HD)
    const float* __restrict__ vcache,  // (NL, KV, S, HD)
    const float* __restrict__ mask,    // (S, L)
    const int*   __restrict__ cp_ptr,
    float* __restrict__ ao)            // (H, HD, L)
{
    extern __shared__ float smem[];
    float* sc     = smem;                 // S_*16
    float* red    = smem + S_ * 16;       // 256
    float* colmax = red + 256;            // 16
    float* colsum = colmax + 16;          // 16

    const int h  = blockIdx.x;
    const int kv = h / G_;
    const int l0 = blockIdx.y * 16;
    const int cp = *cp_ptr;

    const int tid  = threadIdx.x;
    const int lane = tid & 31;
    const int wid  = tid >> 5;
    const int half = lane >> 4;
    const int lx   = lane & 15;
    const int koff = half * 2;

    const float* kbase = kcache + (size_t)(LI_ * KV_ + kv) * S_ * HD_;
    const float* vbase = vcache + (size_t)(LI_ * KV_ + kv) * S_ * HD_;
    const float* knew  = krope + (size_t)kv * L_ * HD_;
    const float* vnew  = vt + (size_t)kv * L_ * HD_;
    const float* qh    = q + (size_t)h * HD_ * L_;

    // ---- Phase 1: scores tile (M=s, N=l), WMMA over d ----
    for (int t = wid; t < S_ / 16; t += 8) {           // wave-uniform: 32 tiles/wave
        const int s0 = t * 16;
        const int srow = s0 + lx;
        const float* arow = (srow >= cp && srow < cp + L_)
                                ? (knew + (size_t)(srow - cp) * HD_)
                                : (kbase + (size_t)srow * HD_);
        v8f acc = {};
#pragma unroll
        for (int k = 0; k < HD_; k += 4) {
            v2f a, b;
            a[0] = arow[k + koff];
            a[1] = arow[k + koff + 1];
            b[0] = qh[(size_t)(k + koff) * L_ + l0 + lx];
            b[1] = qh[(size_t)(k + koff + 1) * L_ + l0 + lx];
            acc = __builtin_amdgcn_wmma_f32_16x16x4_f32(false, a, false, b, (short)0, acc, false, false);
        }
#pragma unroll
        for (int r = 0; r < 8; ++r) {
            const int s = s0 + r + half * 8;
            sc[s * 16 + lx] = acc[r] * SCALE_ + mask[(size_t)s * L_ + l0 + lx];
        }
    }
    __syncthreads();

    // ---- Phase 2: column softmax stats (max & sum); P left unnormalized ----
    {
        const int c = tid & 15;
        const int g = tid >> 4;                        // 0..15
        float m = -3.4e38f;
        for (int r = g; r < S_; r += 16) m = fmaxf(m, sc[r * 16 + c]);
        red[c * 16 + g] = m;
        __syncthreads();
        if (tid < 16) {
            float mm = -3.4e38f;
            for (int gg = 0; gg < 16; ++gg) mm = fmaxf(mm, red[tid * 16 + gg]);
            colmax[tid] = mm;
        }
        __syncthreads();
        const float cm = colmax[c];
        float ssum = 0.0f;
        for (int r = g; r < S_; r += 16) {
            const float e = __expf(sc[r * 16 + c] - cm);
            sc[r * 16 + c] = e;
            ssum += e;
        }
        red[c * 16 + g] = ssum;
        __syncthreads();
        if (tid < 16) {
            float tsum = 0.0f;
            for (int gg = 0; gg < 16; ++gg) tsum += red[tid * 16 + gg];
            colsum[tid] = tsum;
        }
        __syncthreads();
    }

    // ---- Phase 3: O(l,d) = P(l,s) @ V(s,d); each wave owns one 16-wide d slice ----
    {
        const int d0 = wid * 16;
        v8f acc = {};
#pragma unroll 2
        for (int s0 = 0; s0 < S_; s0 += 4) {
            const int sa = s0 + koff;
            const int sb = sa + 1;
            const float* r0 = (sa >= cp && sa < cp + L_)
                                  ? (vnew + (size_t)(sa - cp) * HD_)
                                  : (vbase + (size_t)sa * HD_);
            const float* r1 = (sb >= cp && sb < cp + L_)
                                  ? (vnew + (size_t)(sb - cp) * HD_)
                                  : (vbase + (size_t)sb * HD_);
            v2f a, b;
            a[0] = sc[sa * 16 + lx];     // A = P^T read from LDS (transpose for free)
            a[1] = sc[sb * 16 + lx];
            b[0] = r0[d0 + lx];
            b[1] = r1[d0 + lx];
            acc = __builtin_amdgcn_wmma_f32_16x16x4_f32(false, a, false, b, (short)0, acc, false, false);
        }
#pragma unroll
        for (int r = 0; r < 8; ++r) {
            const int lloc = r + half * 8;
            ao[((size_t)h * HD_ + d0 + lx) * L_ + l0 + lloc] = acc[r] / colsum[lloc];
        }
    }
}

// -------------------------------------------------------------------------
// Host-side launch
// -------------------------------------------------------------------------
extern "C" void kernel_launch(void* const* d_in, const int* in_sizes, int n_in,
                              void* d_out, int out_size, void* d_ws, size_t ws_size,
                              hipStream_t stream)
{
    const float* x      = (const float*)d_in[0];
    const float* cosk   = (const float*)d_in[1];
    const float* sink   = (const float*)d_in[2];
    const float* cost   = (const float*)d_in[3];
    const float* sint   = (const float*)d_in[4];
    const float* mask   = (const float*)d_in[5];
    const float* kcache = (const float*)d_in[6];
    const float* vcache = (const float*)d_in[7];
    const float* wq     = (const float*)d_in[8];
    const float* bq     = (const float*)d_in[9];
    const float* wk     = (const float*)d_in[10];
    const float* bk     = (const float*)d_in[11];
    const float* wv     = (const float*)d_in[12];
    const float* bv     = (const float*)d_in[13];
    const float* wo     = (const float*)d_in[14];
    const int*   cp     = (const int*)d_in[15];

    float* ws   = (float*)d_ws;
    float* qtmp = ws;                                  // H*HD*L   = 1835008
    float* ktmp = qtmp + (size_t)H_ * HD_ * L_;        // KV*HD*L  =  262144
    float* vtmp = ktmp + (size_t)KV_ * HD_ * L_;
    float* kr   = vtmp + (size_t)KV_ * HD_ * L_;
    float* vt   = kr   + (size_t)KV_ * L_ * HD_;
    float* ao   = vt   + (size_t)KV_ * L_ * HD_;       // H*HD*L   = 1835008
    // total 4,718,592 floats = 18 MB of workspace

    const dim3 blk(256);

    // QKV projections (fp32 WMMA GEMMs), block tile 128x128
    gemm_wmma_f32<<<dim3((H_ * HD_) / 128, L_ / 128), blk, 0, stream>>>(wq, bq, x, qtmp, H_ * HD_, D_, L_);
    gemm_wmma_f32<<<dim3((KV_ * HD_) / 128, L_ / 128), blk, 0, stream>>>(wk, bk, x, ktmp, KV_ * HD_, D_, L_);
    gemm_wmma_f32<<<dim3((KV_ * HD_) / 128, L_ / 128), blk, 0, stream>>>(wv, bv, x, vtmp, KV_ * HD_, D_, L_);

    // RoPE + layout changes
    rope_q_kernel<<<(H_ * 64 * L_) / 256, blk, 0, stream>>>(qtmp, cost, sint);
    rope_k_kernel<<<(KV_ * 64 * L_) / 256, blk, 0, stream>>>(ktmp, cosk, sink, kr);
    transpose_v_kernel<<<(KV_ * HD_ * L_) / 256, blk, 0, stream>>>(vtmp, vt);

    // Attention (scores + softmax + P@V), full score column in 256KB LDS
    const size_t lds_bytes = (size_t)(S_ * 16 + 256 + 16 + 16) * sizeof(float);
    attn_kernel<<<dim3(H_, L_ / 16), blk, lds_bytes, stream>>>(qtmp, kr, vt, kcache, vcache, mask, cp, ao);

    // o_proj -> d_out (no bias)
    gemm_wmma_f32<<<dim3(D_ / 128, L_ / 128), blk, 0, stream>>>(wo, nullptr, ao, (float*)d_out, D_, D_, L_);
}